// SpatialAttentionModule_3925600108903
// MI455X (gfx1250) — compile-verified
//
#include <hip/hip_runtime.h>
#include <hip/hip_bf16.h>

// ---------------------------------------------------------------------------
// SpatialAttentionModule on MI455X (gfx1250), wave32, WMMA f32 16x16x4.
//
// Shapes: B=8, N=2048, H=256, HEADS=4, hd=64, BNECK=8, R=B*N=16384.
// attn_reg_loss = exp(log_w) * mean(softmax) = exp(log_w)/N exactly (softmax
// rows sum to 1), so the O(N^2) score path is computed in closed form.
// ---------------------------------------------------------------------------

typedef __attribute__((ext_vector_type(2))) float v2f;
typedef __attribute__((ext_vector_type(8))) float v8f;

#define WMMA_F32(a, b, c) \
    __builtin_amdgcn_wmma_f32_16x16x4_f32(false, (a), false, (b), (short)0, (c), false, false)

__device__ __forceinline__ float elu1(float x) {
    // elu(x)+1 : x>0 ? x+1 : exp(x)
    return x > 0.0f ? x + 1.0f : __expf(x);
}

// ---------------------------------------------------------------------------
// Generic fp32 GEMM: C[M,Ncols] = A[M,K] @ B[K,Ncols] + bias.
// One wave computes one 16x16 tile of C via V_WMMA_F32_16X16X4_F32.
// M multiple of 16, K multiple of 4 (true for all calls). Ragged Ncols
// (24, 8) handled branch-free: clamped index + 0/1 mask multiply, so the
// inner loop has no exec-mask churn and EXEC is all-ones at every WMMA.
// ---------------------------------------------------------------------------
__global__ __launch_bounds__(32) void gemm16_f32(
    const float* __restrict__ A, int lda,
    const float* __restrict__ B, int ldb,
    const float* __restrict__ bias,
    float* __restrict__ C, int ldc,
    int Ncols, int K)
{
    const int tm   = blockIdx.x * 16;
    const int tn   = blockIdx.y * 16;
    const int lane = threadIdx.x & 31;
    const int half = lane >> 4;     // 0: lanes 0-15, 1: lanes 16-31
    const int mr   = lane & 15;
    const int col  = tn + mr;
    const bool cok = (col < Ncols);
    const int colc = cok ? col : (Ncols - 1);   // always-legal address
    const float msk = cok ? 1.0f : 0.0f;

    const float* Arow = A + (size_t)(tm + mr) * lda;
    const float* Bcol = B + colc;

    v8f acc = {};
    for (int k = 0; k < K; k += 4) {
        const int ka = k + 2 * half;
        v2f a, b;
        a.x = Arow[ka];
        a.y = Arow[ka + 1];
        b.x = Bcol[(size_t)ka * ldb] * msk;
        b.y = Bcol[(size_t)(ka + 1) * ldb] * msk;
        acc = WMMA_F32(a, b, acc);
    }

    if (cok) {
        const float bv = bias ? bias[col] : 0.0f;
#pragma unroll
        for (int r = 0; r < 8; ++r) {
            const int row = tm + r + 8 * half;
            C[(size_t)row * ldc + col] = acc[r] + bv;
        }
    }
}

// ---------------------------------------------------------------------------
// zinv[bh*2048 + n] = 1 / (sum_d elu1(k[b,h,n,d]) + 1e-8)
// ---------------------------------------------------------------------------
__global__ __launch_bounds__(256) void z_kernel(
    const float* __restrict__ qkv, float* __restrict__ zinv)
{
    const int idx = blockIdx.x * blockDim.x + threadIdx.x;   // bh*2048 + n
    if (idx >= 8 * 4 * 2048) return;
    const int n  = idx & 2047;
    const int bh = idx >> 11;
    const int b  = bh >> 2;
    const int h  = bh & 3;
    const float* kr = qkv + ((size_t)(b * 2048 + n)) * 768 + 256 + h * 64;
    float s = 0.0f;
#pragma unroll 8
    for (int d = 0; d < 64; ++d) s += elu1(kr[d]);
    zinv[idx] = 1.0f / (s + 1e-8f);
}

// ---------------------------------------------------------------------------
// Partial kv: for chunk c, kvpart[c][bh][d][e] = sum_{n in chunk} kf(n,d)*v(n,e)
// K=2048 split into 8 chunks of 256 -> 4096 waves (vs 512), ~4 waves/SIMD32
// across the 1024 SIMDs, so memory latency is actually hidden.
// grid = (4, 4, 256): z = chunk*32 + bh.
// ---------------------------------------------------------------------------
__global__ __launch_bounds__(32) void kv_partial_kernel(
    const float* __restrict__ qkv, float* __restrict__ kvpart)
{
    const int zc    = blockIdx.z;
    const int chunk = zc >> 5;        // 0..7
    const int bh    = zc & 31;        // 0..31
    const int b  = bh >> 2;
    const int h  = bh & 3;
    const int tm = blockIdx.x * 16;   // d tile
    const int tn = blockIdx.y * 16;   // e tile
    const int lane = threadIdx.x & 31;
    const int half = lane >> 4;
    const int mr   = lane & 15;
    const int d = tm + mr;
    const int e = tn + mr;

    const float* base = qkv + (size_t)b * 2048 * 768;
    const int kcol = 256 + h * 64;
    const int vcol = 512 + h * 64;
    const int nlo = chunk * 256;

    v8f acc = {};
#pragma unroll 4
    for (int n0 = 0; n0 < 256; n0 += 4) {
        const int n = nlo + n0 + 2 * half;
        const float* r0 = base + (size_t)n * 768;
        const float* r1 = r0 + 768;
        v2f a, bb;
        a.x  = elu1(r0[kcol + d]);   // A[d][n] = kf(n,d)
        a.y  = elu1(r1[kcol + d]);
        bb.x = r0[vcol + e];         // B[n][e] = v(n,e)
        bb.y = r1[vcol + e];
        acc = WMMA_F32(a, bb, acc);
    }

    float* out = kvpart + (size_t)zc * 4096;
#pragma unroll
    for (int r = 0; r < 8; ++r)
        out[(tm + r + 8 * half) * 64 + e] = acc[r];
}

// kvmat[bh][i] = sum_{c=0..7} kvpart[c][bh][i]   (deterministic fixed-order)
__global__ __launch_bounds__(256) void kv_reduce_kernel(
    const float* __restrict__ kvpart, float* __restrict__ kvmat)
{
    const int idx = blockIdx.x * blockDim.x + threadIdx.x;   // bh*4096 + i
    if (idx >= 32 * 4096) return;
    float s = 0.0f;
#pragma unroll
    for (int c = 0; c < 8; ++c) s += kvpart[(size_t)c * 32 * 4096 + idx];
    kvmat[idx] = s;
}

// ---------------------------------------------------------------------------
// attn[b,n, h*64+e] = zinv[bh,n] * sum_d elu1(q[b,h,n,d]) * kv[bh][d][e]
// grid = (128, 4, 32); 2048x64x64 GEMM per (b,h), row-scaled epilogue.
// ---------------------------------------------------------------------------
__global__ __launch_bounds__(32) void attnout_kernel(
    const float* __restrict__ qkv, const float* __restrict__ kvmat,
    const float* __restrict__ zinv, float* __restrict__ attn)
{
    const int bh = blockIdx.z;
    const int b  = bh >> 2;
    const int h  = bh & 3;
    const int tm = blockIdx.x * 16;   // n tile
    const int tn = blockIdx.y * 16;   // e tile
    const int lane = threadIdx.x & 31;
    const int half = lane >> 4;
    const int mr   = lane & 15;

    const float* qrow = qkv + ((size_t)(b * 2048 + tm + mr)) * 768 + h * 64;
    const float* Bm   = kvmat + (size_t)bh * 4096;
    const int e = tn + mr;

    v8f acc = {};
#pragma unroll
    for (int k = 0; k < 64; k += 4) {
        const int ka = k + 2 * half;
        v2f a, bb;
        a.x  = elu1(qrow[ka]);        // A[n][d] = qf(n,d)
        a.y  = elu1(qrow[ka + 1]);
        bb.x = Bm[ka * 64 + e];       // B[d][e] = kv(d,e)
        bb.y = Bm[(ka + 1) * 64 + e];
        acc = WMMA_F32(a, bb, acc);
    }

#pragma unroll
    for (int r = 0; r < 8; ++r) {
        const int row = tm + r + 8 * half;            // n
        const float z = zinv[bh * 2048 + row];
        attn[((size_t)(b * 2048 + row)) * 256 + h * 64 + e] = acc[r] * z;
    }
}

// ---------------------------------------------------------------------------
// attn_reg_loss = exp(log_w) / N   (exact: softmax rows sum to 1)
// ---------------------------------------------------------------------------
__global__ void loss_kernel(const float* __restrict__ log_w, float* __restrict__ out)
{
    if (threadIdx.x == 0 && blockIdx.x == 0)
        out[0] = expf(log_w[0]) * (1.0f / 2048.0f);
}

extern "C" void kernel_launch(void* const* d_in, const int* in_sizes, int n_in,
                              void* d_out, int out_size, void* d_ws, size_t ws_size,
                              hipStream_t stream)
{
    (void)in_sizes; (void)n_in; (void)out_size; (void)ws_size;

    const float* x    = (const float*)d_in[0];   // [8,2048,256]
    const float* Wq1  = (const float*)d_in[1];   // [256,24]
    const float* bq1  = (const float*)d_in[2];   // [24]
    const float* Wq2  = (const float*)d_in[3];   // [24,768]
    const float* bq2  = (const float*)d_in[4];   // [768]
    const float* Wo1  = (const float*)d_in[5];   // [256,8]
    const float* bo1  = (const float*)d_in[6];   // [8]
    const float* Wo2  = (const float*)d_in[7];   // [8,256]
    const float* bo2  = (const float*)d_in[8];   // [256]
    // d_in[9] = gu, d_in[10] = gv : unused (loss is closed-form)
    const float* logw = (const float*)d_in[11];  // scalar

    float* out = (float*)d_out;                  // [8,2048,256] ++ [1]

    // workspace layout (floats)
    float* ws     = (float*)d_ws;
    float* t1     = ws;                     // [16384,24]    ->   393216
    float* qkv    = t1     + 393216;        // [16384,768]   -> 12582912
    float* zinv   = qkv    + 12582912;      // [32,2048]     ->    65536
    float* kvmat  = zinv   + 65536;         // [32,64,64]    ->   131072
    float* attn   = kvmat  + 131072;        // [16384,256]   ->  4194304
    float* t2     = attn   + 4194304;       // [16384,8]     ->   131072
    float* kvpart = t2     + 131072;        // [8,32,64,64]  ->  1048576

    // 1) t1 = x @ Wq1 + bq1            [16384,256]x[256,24]
    gemm16_f32<<<dim3(1024, 2), 32, 0, stream>>>(x, 256, Wq1, 24, bq1, t1, 24,
                                                 24, 256);
    // 2) qkv = t1 @ Wq2 + bq2          [16384,24]x[24,768]
    gemm16_f32<<<dim3(1024, 48), 32, 0, stream>>>(t1, 24, Wq2, 768, bq2, qkv, 768,
                                                  768, 24);
    // 3) zinv = 1/(sum_d kf + 1e-8)
    z_kernel<<<256, 256, 0, stream>>>(qkv, zinv);
    // 4a) partial kv = kf^T @ v per (chunk, b, h)
    kv_partial_kernel<<<dim3(4, 4, 256), 32, 0, stream>>>(qkv, kvpart);
    // 4b) fixed-order reduction over the 8 chunks
    kv_reduce_kernel<<<512, 256, 0, stream>>>(kvpart, kvmat);
    // 5) attn = (qf @ kv) * zinv, scattered back to [B,N,H] layout
    attnout_kernel<<<dim3(128, 4, 32), 32, 0, stream>>>(qkv, kvmat, zinv, attn);
    // 6) t2 = attn @ Wo1 + bo1         [16384,256]x[256,8]
    gemm16_f32<<<dim3(1024, 1), 32, 0, stream>>>(attn, 256, Wo1, 8, bo1, t2, 8,
                                                 8, 256);
    // 7) out = t2 @ Wo2 + bo2          [16384,8]x[8,256]
    gemm16_f32<<<dim3(1024, 16), 32, 0, stream>>>(t2, 8, Wo2, 256, bo2, out, 256,
                                                  256, 8);
    // 8) reg loss scalar
    loss_kernel<<<1, 32, 0, stream>>>(logw, out + 4194304);
}